// GlobalContextModule_49134425866252
// MI455X (gfx1250) — compile-verified
//
#include <hip/hip_runtime.h>
#include <hip/hip_bf16.h>

#define D      1024
#define D4     256      // D / 4 (float4 columns)
#define NPART  256      // stage-1 partial blocks
#define LN_EPS 1e-5f

typedef __attribute__((ext_vector_type(2))) float v2f;
typedef __attribute__((ext_vector_type(8))) float v8f;

// ---------------------------------------------------------------------------
// Stage 1: deterministic partial column sums over the concatenation of h0..h3.
// Each block owns one partial slot; thread c owns float4 column c. Streams the
// full 410 MB once; fully coalesced 16B loads.
// ---------------------------------------------------------------------------
__global__ void colsum_stage1(const float* __restrict__ h0, int n0,
                              const float* __restrict__ h1, int n1,
                              const float* __restrict__ h2, int n2,
                              const float* __restrict__ h3, int n3,
                              float* __restrict__ partial) {
    const int c = threadIdx.x;      // 0..255 float4 column
    const int b = blockIdx.x;       // partial slot 0..NPART-1
    float4 acc = make_float4(0.f, 0.f, 0.f, 0.f);
    const float* hs[4] = {h0, h1, h2, h3};
    const int    ns[4] = {n0, n1, n2, n3};
    for (int s = 0; s < 4; ++s) {
        const float4* h4 = (const float4*)hs[s];
        const int n = ns[s];
        for (int r = b; r < n; r += NPART) {
            float4 v = h4[(size_t)r * D4 + c];
            acc.x += v.x; acc.y += v.y; acc.z += v.z; acc.w += v.w;
        }
    }
    ((float4*)partial)[(size_t)b * D4 + c] = acc;
}

// Stage 2: reduce NPART partial rows -> one column-sum vector (fixed order).
__global__ void colsum_stage2(const float* __restrict__ partial,
                              float* __restrict__ colsum) {
    const int col = blockIdx.x * 256 + threadIdx.x;   // 0..1023
    float acc = 0.f;
    for (int p = 0; p < NPART; ++p)
        acc += partial[(size_t)p * D + col];
    colsum[col] = acc;
}

// ---------------------------------------------------------------------------
// Mat-vec y = W*(x*xscale) + bias via V_WMMA_F32_16X16X4_F32.
// One wave (32 threads) per 16-row tile, K marched in steps of 4.
// A layout (16x4 f32): lanes 0-15 -> M=lane, K = k0 + {0,1}; lanes 16-31 ->
// M=lane-16, K = k0 + {2,3}. B (4x16) holds x[k] broadcast across N, so the
// per-lane B value depends only on the same K mapping. C/D: VGPR r holds
// M=r (lanes 0-15) / M=8+r (lanes 16-31); all N columns identical -> lanes 0
// and 16 emit the result.
// ---------------------------------------------------------------------------
__global__ void matvec_wmma(const float* __restrict__ W,
                            const float* __restrict__ bias,
                            const float* __restrict__ x,
                            float xscale,
                            float* __restrict__ y,
                            float* __restrict__ y2) {
    const int lane = threadIdx.x;          // 0..31, single full wave (EXEC all 1s)
    const int half = lane >> 4;            // K-half select
    const int lm   = lane & 15;            // M within tile
    const int m0   = blockIdx.x * 16;
    const float* wrow = W + (size_t)(m0 + lm) * D;

    v8f c = {0.f, 0.f, 0.f, 0.f, 0.f, 0.f, 0.f, 0.f};
    for (int k0 = 0; k0 < D; k0 += 4) {
        const int kb = k0 + 2 * half;
        v2f a, b;
        a[0] = wrow[kb];
        a[1] = wrow[kb + 1];
        b[0] = x[kb]     * xscale;
        b[1] = x[kb + 1] * xscale;
        // (neg_a, A, neg_b, B, c_mod, C, reuse_a, reuse_b)
        c = __builtin_amdgcn_wmma_f32_16x16x4_f32(false, a, false, b,
                                                  (short)0, c, false, false);
    }
    if (lm == 0) {
        const int mbase = m0 + half * 8;
        #pragma unroll
        for (int r = 0; r < 8; ++r) {
            const float v = c[r] + bias[mbase + r];
            y[mbase + r] = v;
            if (y2) y2[mbase + r] = v;
        }
    }
}

// ---------------------------------------------------------------------------
// Fused (h_row + attn) -> LayerNorm -> out. One 256-thread block (8 wave32)
// per row; thread c owns float4 column c. Wave shuffle reduce + 8-slot LDS
// combine. attn/gamma/beta are 4 KB each -> resident in cache.
// ---------------------------------------------------------------------------
__global__ void add_layernorm(const float* __restrict__ h0, int n0,
                              const float* __restrict__ h1, int n1,
                              const float* __restrict__ h2, int n2,
                              const float* __restrict__ h3, int n3,
                              const float* __restrict__ attn,
                              const float* __restrict__ gamma,
                              const float* __restrict__ beta,
                              float* __restrict__ out) {
    __shared__ float s_sum[8];
    __shared__ float s_sq[8];

    int lr = blockIdx.x;                 // row in concat order
    const float* h;
    if (lr < n0)            { h = h0; }
    else { lr -= n0; if (lr < n1) { h = h1; }
    else { lr -= n1; if (lr < n2) { h = h2; }
    else { lr -= n2;               h = h3; } } }

    const int c = threadIdx.x;           // float4 column
    float4 v = ((const float4*)h)[(size_t)lr * D4 + c];
    float4 a = ((const float4*)attn)[c];
    v.x += a.x; v.y += a.y; v.z += a.z; v.w += a.w;

    float s = v.x + v.y + v.z + v.w;
    float q = v.x * v.x + v.y * v.y + v.z * v.z + v.w * v.w;
    #pragma unroll
    for (int m = 16; m >= 1; m >>= 1) {     // wave32 reduce
        s += __shfl_xor(s, m, 32);
        q += __shfl_xor(q, m, 32);
    }
    const int wid = threadIdx.x >> 5;
    if ((threadIdx.x & 31) == 0) { s_sum[wid] = s; s_sq[wid] = q; }
    __syncthreads();
    float ts = 0.f, tq = 0.f;
    #pragma unroll
    for (int w = 0; w < 8; ++w) { ts += s_sum[w]; tq += s_sq[w]; }

    const float mu   = ts * (1.0f / D);
    const float var  = tq * (1.0f / D) - mu * mu;
    const float rstd = rsqrtf(var + LN_EPS);

    float4 g = ((const float4*)gamma)[c];
    float4 bb = ((const float4*)beta)[c];
    float4 o;
    o.x = (v.x - mu) * rstd * g.x + bb.x;
    o.y = (v.y - mu) * rstd * g.y + bb.y;
    o.z = (v.z - mu) * rstd * g.z + bb.z;
    o.w = (v.w - mu) * rstd * g.w + bb.w;
    // out starts with the 1024-float global_vec, then rows in concat order
    ((float4*)out)[(size_t)blockIdx.x * D4 + c + D4] = o;
}

// ---------------------------------------------------------------------------
extern "C" void kernel_launch(void* const* d_in, const int* in_sizes, int n_in,
                              void* d_out, int out_size, void* d_ws, size_t ws_size,
                              hipStream_t stream) {
    const float* h0 = (const float*)d_in[0];  const int n0 = in_sizes[0] / D;
    const float* h1 = (const float*)d_in[1];  const int n1 = in_sizes[1] / D;
    const float* h2 = (const float*)d_in[2];  const int n2 = in_sizes[2] / D;
    const float* h3 = (const float*)d_in[3];  const int n3 = in_sizes[3] / D;
    const float* Wp = (const float*)d_in[4];
    const float* bp = (const float*)d_in[5];
    const float* Wv = (const float*)d_in[6];
    const float* bv = (const float*)d_in[7];
    const float* Wo = (const float*)d_in[8];
    const float* bo = (const float*)d_in[9];
    const float* gamma = (const float*)d_in[10];
    const float* beta  = (const float*)d_in[11];
    float* out = (float*)d_out;

    float* ws      = (float*)d_ws;
    float* partial = ws;                       // NPART * D floats (1 MB)
    float* colsum  = partial + (size_t)NPART * D;
    float* gvec    = colsum + D;
    float* tvec    = gvec + D;
    float* attn    = tvec + D;

    const int total = n0 + n1 + n2 + n3;
    const float invTotal = 1.0f / (float)total;

    // 1) column mean (two-stage, deterministic)
    colsum_stage1<<<NPART, 256, 0, stream>>>(h0, n0, h1, n1, h2, n2, h3, n3, partial);
    colsum_stage2<<<4, 256, 0, stream>>>(partial, colsum);

    // 2) global_vec = Wp*mean + bp  (written to out[0:1024] and ws)
    matvec_wmma<<<64, 32, 0, stream>>>(Wp, bp, colsum, invTotal, gvec, out);
    // 3) t = Wv*g + bv ; attn = Wo*t + bo
    matvec_wmma<<<64, 32, 0, stream>>>(Wv, bv, gvec, 1.0f, tvec, nullptr);
    matvec_wmma<<<64, 32, 0, stream>>>(Wo, bo, tvec, 1.0f, attn, nullptr);

    // 4) fused add + LayerNorm over all rows
    add_layernorm<<<total, 256, 0, stream>>>(h0, n0, h1, n1, h2, n2, h3, n3,
                                             attn, gamma, beta, out);
}